// LightGCNEncoder_53266184405669
// MI455X (gfx1250) — compile-verified
//
#include <hip/hip_runtime.h>
#include <hip/hip_bf16.h>

// LightGCN encoder for MI455X (gfx1250, wave32).
// Memory-bound scatter/gather SpMM; all hot state (2x25.6MB ping-pong,
// 25.6MB accumulator in d_out, 0.8MB degrees) fits in the 192MB L2.
// Finalize pass computes per-row squared norms via V_WMMA_F32_16X16X4_F32
// Gram-matrix trick (free compute on a memory-bound pass).

#define NU   100000
#define NI   100000
#define NN   200000   // NU + NI
#define DIM  32
#define NE   2500000
#define NL   3

typedef __attribute__((ext_vector_type(2))) float v2f;
typedef __attribute__((ext_vector_type(8))) float v8f;

__global__ __launch_bounds__(256) void lgc_init(const float4* __restrict__ ue,
                                                const float4* __restrict__ ie,
                                                float4* __restrict__ xcur,
                                                float4* __restrict__ acc) {
  int i = blockIdx.x * blockDim.x + threadIdx.x;      // over NN*DIM/4
  const int nu4 = NU * DIM / 4;
  const int nt4 = NN * DIM / 4;
  if (i >= nt4) return;
  float4 v = (i < nu4) ? ue[i] : ie[i - nu4];
  xcur[i] = v;
  acc[i]  = v;
}

__global__ __launch_bounds__(256) void lgc_degree(const int* __restrict__ eu,
                                                  const int* __restrict__ ev,
                                                  float* __restrict__ deg) {
  int e = blockIdx.x * blockDim.x + threadIdx.x;
  if (e >= NE) return;
  atomicAdd(&deg[eu[e]], 1.0f);
  atomicAdd(&deg[ev[e] + NU], 1.0f);
}

__global__ __launch_bounds__(256) void lgc_dinv(float* __restrict__ deg) {
  int i = blockIdx.x * blockDim.x + threadIdx.x;
  if (i >= NN) return;
  float d = deg[i];
  deg[i] = (d > 0.0f) ? rsqrtf(d) : 0.0f;
}

// One wave (32 lanes) per undirected edge; lane = embedding dim.
// Handles both directions, reusing the two gathered 128B rows.
__global__ __launch_bounds__(256) void lgc_scatter(const int* __restrict__ eu,
                                                   const int* __restrict__ ev,
                                                   const float* __restrict__ dinv,
                                                   const float* __restrict__ xin,
                                                   float* __restrict__ xout) {
  int wid  = (blockIdx.x * blockDim.x + threadIdx.x) >> 5;
  int lane = threadIdx.x & 31;
  if (wid >= NE) return;
  int u = eu[wid];
  int v = ev[wid] + NU;
  float nrm = dinv[u] * dinv[v];
  float xu = xin[u * DIM + lane];
  float xv = xin[v * DIM + lane];
  atomicAdd(&xout[v * DIM + lane], xu * nrm);   // u -> v
  atomicAdd(&xout[u * DIM + lane], xv * nrm);   // v -> u
}

__global__ __launch_bounds__(256) void lgc_accadd(float4* __restrict__ acc,
                                                  const float4* __restrict__ x) {
  int i = blockIdx.x * blockDim.x + threadIdx.x;  // over NN*DIM/4
  const int nt4 = NN * DIM / 4;
  if (i >= nt4) return;
  float4 a = acc[i];
  float4 b = x[i];
  a.x += b.x; a.y += b.y; a.z += b.z; a.w += b.w;
  acc[i] = a;
}

// One wave per 16x32 row tile. Gram = X * X^T via 8 chained
// v_wmma_f32_16x16x4_f32 (A and B f32 register layouts coincide, so B=A).
// Diagonal of Gram = row squared norms -> L2 normalize (with /4 scaling).
// Grid is sized to EXACTLY NN/16 = 12500 waves so EXEC is all-1s for WMMA.
__global__ __launch_bounds__(320) void lgc_finalize(float* __restrict__ out) {
  int wid  = (blockIdx.x * blockDim.x + threadIdx.x) >> 5;  // 0..12499
  int lane = threadIdx.x & 31;
  float* rows = out + (size_t)wid * 16 * DIM;

  // A-matrix 16x4 f32 layout: lanes 0-15 hold M=lane, K=k0..k0+1;
  // lanes 16-31 hold M=lane-16, K=k0+2..k0+3.
  int m    = lane & 15;
  int koff = (lane >> 4) * 2;

  v8f d = {0.f, 0.f, 0.f, 0.f, 0.f, 0.f, 0.f, 0.f};
#pragma unroll
  for (int c = 0; c < 8; ++c) {
    v2f a = *(const v2f*)(rows + m * DIM + c * 4 + koff);
    // (neg_a, A, neg_b, B, c_mod, C, reuse_a, reuse_b); B = A -> Gram matrix
    d = __builtin_amdgcn_wmma_f32_16x16x4_f32(false, a, false, a,
                                              (short)0, d, false, false);
  }

  // Extract diagonal: row g<8 at (VGPR g, lane g); row g+8 at (VGPR g, lane g+24).
  float inv[16];
#pragma unroll
  for (int g = 0; g < 8; ++g) {
    float s0 = __shfl(d[g], g, 32);
    float s1 = __shfl(d[g], g + 24, 32);
    float n0 = 0.25f * sqrtf(s0);           // ||acc/4||
    float n1 = 0.25f * sqrtf(s1);
    inv[g]     = 0.25f / fmaxf(n0, 1e-12f);
    inv[g + 8] = 0.25f / fmaxf(n1, 1e-12f);
  }

  // out = (acc/4) / max(||acc/4||, 1e-12); coalesced, in-place, per-lane RAW.
#pragma unroll
  for (int r = 0; r < 16; ++r) {
    float val = rows[r * DIM + lane];
    rows[r * DIM + lane] = val * inv[r];
  }
}

extern "C" void kernel_launch(void* const* d_in, const int* in_sizes, int n_in,
                              void* d_out, int out_size, void* d_ws, size_t ws_size,
                              hipStream_t stream) {
  const float* user_emb = (const float*)d_in[0];   // (NU, 32) f32
  const float* item_emb = (const float*)d_in[1];   // (NI, 32) f32
  const int*   eu       = (const int*)d_in[2];     // edge_index row 0
  const int*   ev       = eu + NE;                 // edge_index row 1
  float*       out      = (float*)d_out;           // acc -> (ue|ie), NN*DIM

  // Workspace: deg/dinv (NN), ping-pong x buffers (NN*DIM each) = ~52MB.
  float* deg = (float*)d_ws;
  float* xa  = deg + NN;
  float* xb  = xa + (size_t)NN * DIM;

  const int nt4 = NN * DIM / 4;                    // 1.6M float4 elements

  hipMemsetAsync(deg, 0, (size_t)NN * sizeof(float), stream);
  lgc_init<<<nt4 / 256, 256, 0, stream>>>((const float4*)user_emb,
                                          (const float4*)item_emb,
                                          (float4*)xa, (float4*)out);
  lgc_degree<<<(NE + 255) / 256, 256, 0, stream>>>(eu, ev, deg);
  lgc_dinv<<<(NN + 255) / 256, 256, 0, stream>>>(deg);

  float* xcur = xa;
  float* xnext = xb;
  for (int l = 0; l < NL; ++l) {
    hipMemsetAsync(xnext, 0, (size_t)NN * DIM * sizeof(float), stream);
    // NE waves, 8 waves/block -> exactly 312500 blocks
    lgc_scatter<<<NE / 8, 256, 0, stream>>>(eu, ev, deg, xcur, xnext);
    lgc_accadd<<<nt4 / 256, 256, 0, stream>>>((float4*)out, (const float4*)xnext);
    float* t = xcur; xcur = xnext; xnext = t;      // deterministic host-side swap
  }

  // NN/16 = 12500 waves; 10 waves/block (320 thr) -> exactly 1250 blocks.
  lgc_finalize<<<1250, 320, 0, stream>>>(out);
}